// xGAT_16028817949315
// MI455X (gfx1250) — compile-verified
//
#include <hip/hip_runtime.h>
#include <hip/hip_bf16.h>

typedef __attribute__((ext_vector_type(2))) float v2f;
typedef __attribute__((ext_vector_type(8))) float v8f;

#define NEG_SLOPE 0.2f

// ---------- monotone float <-> uint encoding for atomicMax-based segment max ----------
__device__ __forceinline__ unsigned enc_f(float f) {
  unsigned u = __float_as_uint(f);
  return (u & 0x80000000u) ? ~u : (u | 0x80000000u);
}
__device__ __forceinline__ float dec_f(unsigned u) {
  return __uint_as_float((u & 0x80000000u) ? (u & 0x7fffffffu) : ~u);
}

// ---------- utility ----------
__global__ void k_zero(float* __restrict__ p, long n) {
  long i = blockIdx.x * (long)blockDim.x + threadIdx.x;
  if (i < n) p[i] = 0.0f;
}

// ---------- edge-attr mean per destination (fill_value='mean' for self loops) ----------
__global__ void k_edge_ma(const float* __restrict__ eattr, const int* __restrict__ dst,
                          float* __restrict__ cnt, float* __restrict__ ma, int E) {
  long i = blockIdx.x * (long)blockDim.x + threadIdx.x;
  if (i >= (long)E * 16) return;
  int e = (int)(i >> 4), j = (int)(i & 15);
  int d = dst[e];
  atomicAdd(ma + (size_t)d * 16 + j, eattr[i]);
  if (j == 0) atomicAdd(cnt + d, 1.0f);
}

__global__ void k_ma_div(float* __restrict__ ma, const float* __restrict__ cnt, int N) {
  long i = blockIdx.x * (long)blockDim.x + threadIdx.x;
  if (i >= (long)N * 16) return;
  ma[i] /= fmaxf(cnt[i >> 4], 1.0f);
}

// ---------- fp32 WMMA GEMM: Y[nrows,OUTD] = X[nrows,K] @ W[K,OUTD] + bias ----------
// one wave = 16 rows x 64 cols (4 accumulators); K in steps of 4 via v_wmma_f32_16x16x4_f32
__global__ void k_gemm_bias(const float* __restrict__ X, const float* __restrict__ W,
                            const float* __restrict__ bias, float* __restrict__ Y,
                            int nrows, int K, int OUTD) {
  int lane = threadIdx.x & 31;
  int tm = blockIdx.x * 4 + (threadIdx.x >> 5);
  int m0 = tm * 16;
  if (m0 >= nrows) return;
  int n0 = blockIdx.y * 64;
  int row = m0 + (lane & 15);
  int kk  = (lane >> 4) * 2;           // lanes 0-15: K 0,1 ; lanes 16-31: K 2,3
  int coll = n0 + (lane & 15);
  v8f acc0 = {}, acc1 = {}, acc2 = {}, acc3 = {};
  const float* xp = X + (size_t)row * K;
  for (int k = 0; k < K; k += 4) {
    v2f a; a.x = xp[k + kk]; a.y = xp[k + kk + 1];
    const float* w0 = W + (size_t)(k + kk) * OUTD + coll;
    const float* w1 = w0 + OUTD;
    v2f b0; b0.x = w0[0];  b0.y = w1[0];
    v2f b1; b1.x = w0[16]; b1.y = w1[16];
    v2f b2; b2.x = w0[32]; b2.y = w1[32];
    v2f b3; b3.x = w0[48]; b3.y = w1[48];
    acc0 = __builtin_amdgcn_wmma_f32_16x16x4_f32(false, a, false, b0, (short)0, acc0, false, false);
    acc1 = __builtin_amdgcn_wmma_f32_16x16x4_f32(false, a, false, b1, (short)0, acc1, false, false);
    acc2 = __builtin_amdgcn_wmma_f32_16x16x4_f32(false, a, false, b2, (short)0, acc2, false, false);
    acc3 = __builtin_amdgcn_wmma_f32_16x16x4_f32(false, a, false, b3, (short)0, acc3, false, false);
  }
  int rb = m0 + ((lane >> 4) << 3);    // C/D layout: lanes>=16 hold M = v+8
  float bi0 = bias[coll], bi1 = bias[coll + 16], bi2 = bias[coll + 32], bi3 = bias[coll + 48];
#pragma unroll
  for (int v = 0; v < 8; ++v) {
    float* yp = Y + (size_t)(rb + v) * OUTD + coll;
    yp[0]  = acc0[v] + bi0;
    yp[16] = acc1[v] + bi1;
    yp[32] = acc2[v] + bi2;
    yp[48] = acc3[v] + bi3;
  }
}

// ---------- attention logits + per-(dst,head) running max; one wave per (virtual) edge ----------
__global__ void k_logits(const float* __restrict__ xl, const float* __restrict__ xr,
                         const float* __restrict__ eattr, const float* __restrict__ ma,
                         const float* __restrict__ We, const float* __restrict__ att,
                         const int* __restrict__ src, const int* __restrict__ dst,
                         float* __restrict__ pbuf, unsigned* __restrict__ maxb,
                         int E, int E2, int OUTD) {
  long gt = blockIdx.x * (long)blockDim.x + threadIdx.x;
  int wid = (int)(gt >> 5);
  int lane = threadIdx.x & 31;
  if (wid >= E2) return;
  int s, d; const float* ea;
  if (wid < E) { s = src[wid]; d = dst[wid]; ea = eattr + (size_t)wid * 16; }
  else         { s = wid - E;  d = s;        ea = ma    + (size_t)s   * 16; }
  const int CPT = OUTD >> 5;           // channels per lane: 4 (OUTD=128) or 8 (OUTD=256)
  int c = lane * CPT;
  int head = lane >> 3;                // 8 lanes per head
  float eav = ea[lane & 15];
  const float* pxl = xl + (size_t)s * OUTD + c;
  const float* pxr = xr + (size_t)d * OUTD + c;
  const float* pat = att + c;          // att flat [4, CH] == OUTD contiguous
  float partial = 0.0f;
  for (int j = 0; j < CPT; j += 4) {
    float4 vl = *(const float4*)(pxl + j);
    float4 vr = *(const float4*)(pxr + j);
    float z0 = vl.x + vr.x, z1 = vl.y + vr.y, z2 = vl.z + vr.z, z3 = vl.w + vr.w;
#pragma unroll
    for (int k = 0; k < 16; ++k) {
      float ak = __shfl(eav, k, 32);
      float4 w = *(const float4*)(We + (size_t)k * OUTD + c + j);
      z0 = fmaf(ak, w.x, z0); z1 = fmaf(ak, w.y, z1);
      z2 = fmaf(ak, w.z, z2); z3 = fmaf(ak, w.w, z3);
    }
    z0 = z0 >= 0.0f ? z0 : NEG_SLOPE * z0;
    z1 = z1 >= 0.0f ? z1 : NEG_SLOPE * z1;
    z2 = z2 >= 0.0f ? z2 : NEG_SLOPE * z2;
    z3 = z3 >= 0.0f ? z3 : NEG_SLOPE * z3;
    float4 a4 = *(const float4*)(pat + j);
    partial = fmaf(z0, a4.x, partial);
    partial = fmaf(z1, a4.y, partial);
    partial = fmaf(z2, a4.z, partial);
    partial = fmaf(z3, a4.w, partial);
  }
  partial += __shfl_xor(partial, 1, 32);
  partial += __shfl_xor(partial, 2, 32);
  partial += __shfl_xor(partial, 4, 32);
  if ((lane & 7) == 0) {
    pbuf[(size_t)wid * 4 + head] = partial;
    atomicMax(maxb + (size_t)d * 4 + head, enc_f(partial));
  }
}

// ---------- exp(logit - max) + per-(dst,head) sum ----------
__global__ void k_expsum(const int* __restrict__ dst, float* __restrict__ pbuf,
                         const unsigned* __restrict__ maxb, float* __restrict__ sumb,
                         int E, int E2) {
  long i = blockIdx.x * (long)blockDim.x + threadIdx.x;
  if (i >= (long)E2 * 4) return;
  int h = (int)(i & 3);
  int e = (int)(i >> 2);
  int d = (e < E) ? dst[e] : (e - E);
  float m = dec_f(maxb[(size_t)d * 4 + h]);
  float p = expf(pbuf[i] - m);
  pbuf[i] = p;
  atomicAdd(sumb + (size_t)d * 4 + h, p);
}

// ---------- alpha-weighted scatter-add of source features; one wave per edge ----------
__global__ void k_aggregate(const float* __restrict__ xl, const float* __restrict__ pbuf,
                            const float* __restrict__ sumb,
                            const int* __restrict__ src, const int* __restrict__ dst,
                            float* __restrict__ agg, int E, int E2, int OUTD) {
  long gt = blockIdx.x * (long)blockDim.x + threadIdx.x;
  int wid = (int)(gt >> 5);
  int lane = threadIdx.x & 31;
  if (wid >= E2) return;
  int s, d;
  if (wid < E) { s = src[wid]; d = dst[wid]; }
  else         { s = wid - E;  d = s; }
  const int CPT = OUTD >> 5;
  int c = lane * CPT;
  int head = lane >> 3;
  float alpha = pbuf[(size_t)wid * 4 + head] / (sumb[(size_t)d * 4 + head] + 1e-16f);
  const float* pxl = xl + (size_t)s * OUTD + c;
  float* pag = agg + (size_t)d * OUTD + c;
#pragma unroll
  for (int j = 0; j < 8; ++j) {
    if (j < CPT) atomicAdd(pag + j, alpha * pxl[j]);
  }
}

// ---------- epilogues ----------
__global__ void k_epi_concat(const float* __restrict__ agg, const float* __restrict__ bias,
                             float* __restrict__ h, long total, int OUTD) {
  long i = blockIdx.x * (long)blockDim.x + threadIdx.x;
  if (i >= total) return;
  int c = (int)(i & (OUTD - 1));
  h[i] = fmaxf(agg[i] + bias[c], 0.0f);
}

__global__ void k_epi_mean(const float* __restrict__ agg, const float* __restrict__ bias,
                           float* __restrict__ h, int N) {
  long i = blockIdx.x * (long)blockDim.x + threadIdx.x;
  if (i >= (long)N * 64) return;
  long n = i >> 6; int c = (int)(i & 63);
  const float* a = agg + n * 256 + c;
  float v = (a[0] + a[64] + a[128] + a[192]) * 0.25f + bias[c];
  h[i] = fmaxf(v, 0.0f);
}

// ---------- global mean pool + final linear ----------
__global__ void k_pool(const float* __restrict__ h3, const int* __restrict__ batch,
                       float* __restrict__ gsum, float* __restrict__ gcnt, int N) {
  long i = blockIdx.x * (long)blockDim.x + threadIdx.x;
  if (i >= (long)N * 64) return;
  int n = (int)(i >> 6), c = (int)(i & 63);
  int b = batch[n];
  atomicAdd(gsum + (size_t)b * 64 + c, h3[i]);
  if (c == 0) atomicAdd(gcnt + b, 1.0f);
}

__global__ void k_final(const float* __restrict__ gsum, const float* __restrict__ gcnt,
                        const float* __restrict__ Wlin, const float* __restrict__ blin,
                        float* __restrict__ out, int G) {
  int i = blockIdx.x * blockDim.x + threadIdx.x;
  if (i >= G * 16) return;
  int g = i >> 4, o = i & 15;
  float inv = 1.0f / fmaxf(gcnt[g], 1.0f);
  float acc = blin[o];
  for (int c = 0; c < 64; ++c)
    acc = fmaf(gsum[(size_t)g * 64 + c] * inv, Wlin[c * 16 + o], acc);
  out[i] = acc;
}

extern "C" void kernel_launch(void* const* d_in, const int* in_sizes, int n_in,
                              void* d_out, int out_size, void* d_ws, size_t ws_size,
                              hipStream_t stream) {
  const float* x     = (const float*)d_in[0];
  const int*   eidx  = (const int*)  d_in[1];
  const int*   batch = (const int*)  d_in[2];
  const float* eattr = (const float*)d_in[3];
  const float *Wl0=(const float*)d_in[4],  *bl0=(const float*)d_in[5];
  const float *Wr0=(const float*)d_in[6],  *br0=(const float*)d_in[7];
  const float *We0=(const float*)d_in[8],  *att0=(const float*)d_in[9], *bo0=(const float*)d_in[10];
  const float *Wlh=(const float*)d_in[11], *blh=(const float*)d_in[12];
  const float *Wrh=(const float*)d_in[13], *brh=(const float*)d_in[14];
  const float *Weh=(const float*)d_in[15], *atth=(const float*)d_in[16], *boh=(const float*)d_in[17];
  const float *Wlf=(const float*)d_in[18], *blf=(const float*)d_in[19];
  const float *Wrf=(const float*)d_in[20], *brf=(const float*)d_in[21];
  const float *Wef=(const float*)d_in[22], *attf=(const float*)d_in[23], *bof=(const float*)d_in[24];
  const float *Wlin=(const float*)d_in[25],*blin=(const float*)d_in[26];

  const int N  = in_sizes[2];          // batch has N elements
  const int E  = in_sizes[1] / 2;
  const int E2 = E + N;                // edges + self loops
  const int G  = out_size / 16;
  const int* src = eidx;
  const int* dst = eidx + E;

  float* ws = (float*)d_ws;
  size_t off = 0;
  float* cnt  = ws + off; off += (size_t)N;
  float* ma   = ws + off; off += (size_t)N * 16;
  float* xl   = ws + off; off += (size_t)N * 256;
  float* xr   = ws + off; off += (size_t)N * 256;   // also reused as scatter accumulator
  float* hbuf = ws + off; off += (size_t)N * 128;   // ping-pong hidden state (and final h3)
  float* pbuf = ws + off; off += (size_t)E2 * 4;
  unsigned* maxb = (unsigned*)(ws + off); off += (size_t)N * 4;
  float* sumb = ws + off; off += (size_t)N * 4;
  float* gsum = ws + off; off += (size_t)G * 64;
  float* gcnt = ws + off; off += (size_t)G;

  auto zero = [&](float* p, long n) {
    k_zero<<<dim3((unsigned)((n + 255) / 256)), dim3(256), 0, stream>>>(p, n);
  };

  // edge-attr mean per destination (shared by all layers)
  zero(cnt, N);
  zero(ma, (long)N * 16);
  k_edge_ma<<<dim3((unsigned)(((long)E * 16 + 255) / 256)), dim3(256), 0, stream>>>(eattr, dst, cnt, ma, E);
  k_ma_div<<<dim3((unsigned)(((long)N * 16 + 255) / 256)), dim3(256), 0, stream>>>(ma, cnt, N);

  auto layer = [&](const float* X, int K, int OUTD,
                   const float* Wl, const float* bl, const float* Wr, const float* br,
                   const float* We, const float* att, const float* bo, bool concat) {
    int tiles_m = (N + 15) / 16;
    dim3 gg((unsigned)((tiles_m + 3) / 4), (unsigned)(OUTD / 64));
    k_gemm_bias<<<gg, 128, 0, stream>>>(X, Wl, bl, xl, N, K, OUTD);
    k_gemm_bias<<<gg, 128, 0, stream>>>(X, Wr, br, xr, N, K, OUTD);
    zero((float*)maxb, (long)N * 4);
    zero(sumb, (long)N * 4);
    long nthr = (long)E2 * 32;
    k_logits<<<dim3((unsigned)((nthr + 255) / 256)), dim3(256), 0, stream>>>(
        xl, xr, eattr, ma, We, att, src, dst, pbuf, maxb, E, E2, OUTD);
    k_expsum<<<dim3((unsigned)(((long)E2 * 4 + 255) / 256)), dim3(256), 0, stream>>>(
        dst, pbuf, maxb, sumb, E, E2);
    zero(xr, (long)N * OUTD);          // xr now reused as the aggregation buffer
    k_aggregate<<<dim3((unsigned)((nthr + 255) / 256)), dim3(256), 0, stream>>>(
        xl, pbuf, sumb, src, dst, xr, E, E2, OUTD);
    if (concat)
      k_epi_concat<<<dim3((unsigned)(((long)N * OUTD + 255) / 256)), dim3(256), 0, stream>>>(
          xr, bo, hbuf, (long)N * OUTD, OUTD);
    else
      k_epi_mean<<<dim3((unsigned)(((long)N * 64 + 255) / 256)), dim3(256), 0, stream>>>(
          xr, bo, hbuf, N);
  };

  layer(x,    128, 128, Wl0, bl0, Wr0, br0, We0, att0, bo0, true);
  layer(hbuf, 128, 128, Wlh, blh, Wrh, brh, Weh, atth, boh, true);
  layer(hbuf, 128, 256, Wlf, blf, Wrf, brf, Wef, attf, bof, false);

  zero(gsum, (long)G * 64);
  zero(gcnt, (long)G);
  k_pool<<<dim3((unsigned)(((long)N * 64 + 255) / 256)), dim3(256), 0, stream>>>(hbuf, batch, gsum, gcnt, N);
  k_final<<<dim3((unsigned)((G * 16 + 255) / 256)), dim3(256), 0, stream>>>(gsum, gcnt, Wlin, blin, (float*)d_out, G);
}